// HeteroAttnConv_40956808134667
// MI455X (gfx1250) — compile-verified
//
#include <hip/hip_runtime.h>
#include <math.h>

#define N_NODES 10000
#define N_EDGES 15000
#define DIM 64
#define HEADS 4
#define LN_EPS 1e-5f
#define WPB 8            // waves per block (wave32)
#define TPB (WPB * 32)

typedef __attribute__((ext_vector_type(2))) float v2f;
typedef __attribute__((ext_vector_type(8))) float v8f;

// ---------------------------------------------------------------------------
// CDNA5 WMMA: D(16x16,f32) = A(16x4,f32) x B(4x16,f32) + C
// ---------------------------------------------------------------------------
__device__ __forceinline__ v8f wmma_f32_k4(v2f a, v2f b, v8f c) {
  return __builtin_amdgcn_wmma_f32_16x16x4_f32(false, a, false, b, (short)0, c,
                                               false, false);
}

__device__ __forceinline__ void lds_fence() {
  __asm__ volatile("s_wait_dscnt 0x0" ::: "memory");
}

// B-matrix (4x16) tile of W^T: B[k][n] = W[n0+n][4t+k].
// lane L -> n = L&15, k-pair = 2*(L>>4); one b64 load per lane per tile.
// Weights are single-use streaming data -> non-temporal load (TH=NT) so the
// 1.15 GB weight stream does not evict the reusable feat/query set from L2.
__device__ __forceinline__ v2f load_w_tile(const float* __restrict__ W, int n0,
                                           int t, int lane) {
  const int row = n0 + (lane & 15);
  const int col = 4 * t + ((lane >> 4) << 1);
  return __builtin_nontemporal_load((const v2f*)(W + row * DIM + col));
}

// Monotone uint encoding of float for order-invariant atomic max.
__device__ __forceinline__ unsigned enc_f32(float f) {
  unsigned u = __float_as_uint(f);
  return (u & 0x80000000u) ? ~u : (u | 0x80000000u);
}
__device__ __forceinline__ float dec_f32(unsigned u) {
  return (u & 0x80000000u) ? __uint_as_float(u & 0x7FFFFFFFu)
                           : __uint_as_float(~u);
}

// ---------------------------------------------------------------------------
// Kernel 0: init workspace accumulators
// ---------------------------------------------------------------------------
__global__ void init_kernel(unsigned* __restrict__ smaxkey,
                            float* __restrict__ denom,
                            float* __restrict__ agg) {
  const int idx = blockIdx.x * blockDim.x + threadIdx.x;
  if (idx < N_NODES * DIM) agg[idx] = 0.0f;
  if (idx < N_NODES * HEADS) {
    denom[idx] = 0.0f;
    smaxkey[idx] = 0u;  // below enc(-FLT_MAX); always overwritten before decode
  }
}

// ---------------------------------------------------------------------------
// Kernel 1: per-edge K/V matvecs (WMMA fp32), relu, v out, per-head scores,
//           atomic segment-max of scores per dst node.  One wave per edge.
// ---------------------------------------------------------------------------
__global__ __launch_bounds__(TPB) void edge_kernel(
    const float* __restrict__ feat, const float* __restrict__ query,
    const float* __restrict__ skw, const float* __restrict__ dkw,
    const float* __restrict__ skb, const float* __restrict__ dkb,
    const float* __restrict__ svw, const float* __restrict__ dvw,
    const float* __restrict__ svb, const float* __restrict__ dvb,
    const int* __restrict__ src, const int* __restrict__ dst,
    float* __restrict__ v_out, float* __restrict__ score,
    unsigned* __restrict__ smaxkey) {
  __shared__ float lds[WPB * 320];
  const int wid = threadIdx.x >> 5;
  const int lane = threadIdx.x & 31;
  const int e = blockIdx.x * WPB + wid;
  if (e >= N_EDGES) return;  // wave-uniform; EXEC stays all-ones for WMMA

  float* fu = &lds[wid * 320];
  float* fv = fu + 64;
  float* qb = fu + 128;
  float* kb = fu + 192;
  float* vb = fu + 256;

  const int s = src[e];
  const int d = dst[e];
  fu[lane]      = feat[s * DIM + lane];
  fu[lane + 32] = feat[s * DIM + 32 + lane];
  fv[lane]      = feat[d * DIM + lane];
  fv[lane + 32] = feat[d * DIM + 32 + lane];
  qb[lane]      = query[d * DIM + lane];
  qb[lane + 32] = query[d * DIM + 32 + lane];
  lds_fence();

  // A tiles: broadcast feature chunks. A[m][k] = f[4t+k] for all rows m.
  v2f afu[16], afv[16];
  const int c2 = (lane >> 4) << 1;
#pragma unroll
  for (int t = 0; t < 16; ++t) {
    afu[t] = *(const v2f*)(fu + 4 * t + c2);
    afv[t] = *(const v2f*)(fv + 4 * t + c2);
  }

  const size_t ebase = (size_t)e * (DIM * DIM);
  const float* Wks = skw + ebase;
  const float* Wkd = dkw + ebase;
  const float* Wvs = svw + ebase;
  const float* Wvd = dvw + ebase;
  const int bidx = e * DIM + (lane & 15);

  // K = relu(Wks*fu + skb + Wkd*fv + dkb); D row 0 = out[n0+lane], lanes 0-15
  for (int n0 = 0; n0 < DIM; n0 += 16) {
    v8f acc = {0.f, 0.f, 0.f, 0.f, 0.f, 0.f, 0.f, 0.f};
#pragma unroll
    for (int t = 0; t < 16; ++t)
      acc = wmma_f32_k4(afu[t], load_w_tile(Wks, n0, t, lane), acc);
#pragma unroll
    for (int t = 0; t < 16; ++t)
      acc = wmma_f32_k4(afv[t], load_w_tile(Wkd, n0, t, lane), acc);
    const float r =
        fmaxf(acc[0] + __builtin_nontemporal_load(skb + bidx + n0) +
                  __builtin_nontemporal_load(dkb + bidx + n0),
              0.0f);
    if (lane < 16) kb[n0 + lane] = r;
  }
  // V = relu(Wvs*fu + svb + Wvd*fv + dvb)
  for (int n0 = 0; n0 < DIM; n0 += 16) {
    v8f acc = {0.f, 0.f, 0.f, 0.f, 0.f, 0.f, 0.f, 0.f};
#pragma unroll
    for (int t = 0; t < 16; ++t)
      acc = wmma_f32_k4(afu[t], load_w_tile(Wvs, n0, t, lane), acc);
#pragma unroll
    for (int t = 0; t < 16; ++t)
      acc = wmma_f32_k4(afv[t], load_w_tile(Wvd, n0, t, lane), acc);
    const float r =
        fmaxf(acc[0] + __builtin_nontemporal_load(svb + bidx + n0) +
                  __builtin_nontemporal_load(dvb + bidx + n0),
              0.0f);
    if (lane < 16) vb[n0 + lane] = r;
  }
  lds_fence();

  // v output (E, D)
  v_out[e * DIM + lane]      = vb[lane];
  v_out[e * DIM + 32 + lane] = vb[lane + 32];

  // score[e][h] = dot(k_h, q_h); head = element/16.
  float p1 = kb[lane] * qb[lane];            // heads 0 (lanes0-15), 1 (16-31)
  float p2 = kb[lane + 32] * qb[lane + 32];  // heads 2, 3
#pragma unroll
  for (int off = 8; off >= 1; off >>= 1) {
    p1 += __shfl_xor(p1, off, 32);
    p2 += __shfl_xor(p2, off, 32);
  }
  if ((lane & 15) == 0) {
    const int g = lane >> 4;  // 0 or 1
    score[e * HEADS + g] = p1;
    score[e * HEADS + 2 + g] = p2;
    atomicMax(&smaxkey[d * HEADS + g], enc_f32(p1));
    atomicMax(&smaxkey[d * HEADS + 2 + g], enc_f32(p2));
  }
}

// ---------------------------------------------------------------------------
// Kernel 2: ex = exp(score - smax[dst]); denom[dst] += ex
// ---------------------------------------------------------------------------
__global__ void softmax_kernel(const float* __restrict__ score,
                               const unsigned* __restrict__ smaxkey,
                               const int* __restrict__ dst,
                               float* __restrict__ exv,
                               float* __restrict__ denom) {
  const int idx = blockIdx.x * blockDim.x + threadIdx.x;
  if (idx >= N_EDGES * HEADS) return;
  const int e = idx >> 2, h = idx & 3;
  const int d = dst[e];
  const float ex = expf(score[idx] - dec_f32(smaxkey[d * HEADS + h]));
  exv[idx] = ex;
  atomicAdd(&denom[d * HEADS + h], ex);
}

// ---------------------------------------------------------------------------
// Kernel 3: attn = ex/denom[dst]; scatter v*attn into agg[dst]
// ---------------------------------------------------------------------------
__global__ void scatter_kernel(const float* __restrict__ exv,
                               const float* __restrict__ denom,
                               const float* __restrict__ v_out,
                               const int* __restrict__ dst,
                               float* __restrict__ attn_out,
                               float* __restrict__ agg) {
  const int idx = blockIdx.x * blockDim.x + threadIdx.x;
  if (idx >= N_EDGES * DIM) return;
  const int e = idx >> 6, i = idx & 63, h = i >> 4;
  const int d = dst[e];
  const float a = exv[e * HEADS + h] / denom[d * HEADS + h];
  atomicAdd(&agg[d * DIM + i], v_out[idx] * a);
  if (i < HEADS) attn_out[e * HEADS + i] = exv[e * HEADS + i] / denom[d * HEADS + i];
}

// ---------------------------------------------------------------------------
// Kernel 4: per-node matvec (WMMA) + bias + relu + residual + layernorm.
// One wave per node.
// ---------------------------------------------------------------------------
__global__ __launch_bounds__(TPB) void node_kernel(
    const float* __restrict__ feat, const float* __restrict__ nw,
    const float* __restrict__ nb, const float* __restrict__ gamma,
    const float* __restrict__ beta, const float* __restrict__ agg,
    float* __restrict__ nf_out) {
  __shared__ float lds[WPB * 128];
  const int wid = threadIdx.x >> 5;
  const int lane = threadIdx.x & 31;
  const int n = blockIdx.x * WPB + wid;
  if (n >= N_NODES) return;  // wave-uniform

  float* ab = &lds[wid * 128];
  float* yb = ab + 64;
  ab[lane]      = agg[n * DIM + lane];
  ab[lane + 32] = agg[n * DIM + 32 + lane];
  lds_fence();

  v2f at[16];
  const int c2 = (lane >> 4) << 1;
#pragma unroll
  for (int t = 0; t < 16; ++t) at[t] = *(const v2f*)(ab + 4 * t + c2);

  const float* W = nw + (size_t)n * (DIM * DIM);
  const int ridx = n * DIM + (lane & 15);
  for (int n0 = 0; n0 < DIM; n0 += 16) {
    v8f acc = {0.f, 0.f, 0.f, 0.f, 0.f, 0.f, 0.f, 0.f};
#pragma unroll
    for (int t = 0; t < 16; ++t)
      acc = wmma_f32_k4(at[t], load_w_tile(W, n0, t, lane), acc);
    const float y =
        fmaxf(acc[0] + __builtin_nontemporal_load(nb + ridx + n0), 0.0f) +
        feat[ridx + n0];  // relu + residual
    if (lane < 16) yb[n0 + lane] = y;
  }
  lds_fence();

  const float y0 = yb[lane], y1 = yb[lane + 32];
  float s = y0 + y1, sq = y0 * y0 + y1 * y1;
#pragma unroll
  for (int off = 16; off >= 1; off >>= 1) {
    s += __shfl_xor(s, off, 32);
    sq += __shfl_xor(sq, off, 32);
  }
  const float mu = s * (1.0f / DIM);
  const float var = sq * (1.0f / DIM) - mu * mu;
  const float rs = rsqrtf(var + LN_EPS);
  nf_out[n * DIM + lane]      = (y0 - mu) * rs * gamma[lane] + beta[lane];
  nf_out[n * DIM + 32 + lane] = (y1 - mu) * rs * gamma[lane + 32] + beta[lane + 32];
}

// ---------------------------------------------------------------------------
extern "C" void kernel_launch(void* const* d_in, const int* in_sizes, int n_in,
                              void* d_out, int out_size, void* d_ws,
                              size_t ws_size, hipStream_t stream) {
  const float* feat  = (const float*)d_in[0];
  const float* query = (const float*)d_in[1];
  const float* nw    = (const float*)d_in[2];
  const float* nb    = (const float*)d_in[3];
  const float* skw   = (const float*)d_in[4];
  const float* dkw   = (const float*)d_in[5];
  const float* skb   = (const float*)d_in[6];
  const float* dkb   = (const float*)d_in[7];
  const float* svw   = (const float*)d_in[8];
  const float* dvw   = (const float*)d_in[9];
  const float* svb   = (const float*)d_in[10];
  const float* dvb   = (const float*)d_in[11];
  const float* gamma = (const float*)d_in[12];
  const float* beta  = (const float*)d_in[13];
  const int*   src   = (const int*)d_in[14];
  const int*   dst   = (const int*)d_in[15];

  // d_out: node_feat (N*D) | v (E*D) | attn (E*H)
  float* nf_out   = (float*)d_out;
  float* v_out    = nf_out + (size_t)N_NODES * DIM;
  float* attn_out = v_out + (size_t)N_EDGES * DIM;

  // workspace: smaxkey | denom | score | ex | agg  (~3.4 MB total)
  unsigned* smaxkey = (unsigned*)d_ws;
  float* denom = (float*)d_ws + N_NODES * HEADS;
  float* score = denom + N_NODES * HEADS;
  float* exv   = score + N_EDGES * HEADS;
  float* agg   = exv + N_EDGES * HEADS;

  init_kernel<<<(N_NODES * DIM + 255) / 256, 256, 0, stream>>>(smaxkey, denom,
                                                               agg);
  edge_kernel<<<N_EDGES / WPB, TPB, 0, stream>>>(
      feat, query, skw, dkw, skb, dkb, svw, dvw, svb, dvb, src, dst, v_out,
      score, smaxkey);
  softmax_kernel<<<(N_EDGES * HEADS + 255) / 256, 256, 0, stream>>>(
      score, smaxkey, dst, exv, denom);
  scatter_kernel<<<(N_EDGES * DIM + 255) / 256, 256, 0, stream>>>(
      exv, denom, v_out, dst, attn_out, agg);
  node_kernel<<<N_NODES / WPB, TPB, 0, stream>>>(feat, nw, nb, gamma, beta,
                                                 agg, nf_out);
}